// Net_23587960389983
// MI455X (gfx1250) — compile-verified
//
#include <hip/hip_runtime.h>
#include <math.h>

#define F_IN  1433
#define F_HID 16
#define F_OUT 7
#define KG    359   // ceil(F_IN/4) k-groups of 4 (last one padded with zeros in Wp)

typedef __attribute__((ext_vector_type(2))) float v2f;
typedef __attribute__((ext_vector_type(8))) float v8f;

__device__ __forceinline__ void atomAddF(float* p, float v) {
    (void)__hip_atomic_fetch_add(p, v, __ATOMIC_RELAXED, __HIP_MEMORY_SCOPE_AGENT);
}

// ---------------- degree / normalization ----------------
__global__ void deg_init_kernel(float* __restrict__ deg, int n) {
    int i = blockIdx.x * blockDim.x + threadIdx.x;
    if (i < n) deg[i] = 1.0f;                 // self-loop contributes 1
}

__global__ void deg_edge_kernel(const int* __restrict__ dst, float* __restrict__ deg, int e) {
    int i = blockIdx.x * blockDim.x + threadIdx.x;
    if (i < e) atomAddF(&deg[dst[i]], 1.0f);
}

__global__ void deg_rsqrt_kernel(float* __restrict__ deg, int n) {
    int i = blockIdx.x * blockDim.x + threadIdx.x;
    if (i < n) {
        float d = deg[i];
        deg[i] = (d > 0.0f) ? rsqrtf(d) : 0.0f;   // dis = deg^{-1/2}
    }
}

// ---------------- W1 repack into WMMA B-lane order ----------------
// Wp[g*64 + lane*2 + j] = W1[(4g + 2*(lane>>4) + j)*16 + (lane&15)], zero past K.
// A wave's B fragment for k-group g is then one coalesced b64 load: ((v2f*)Wp)[g*32+lane].
__global__ void pack_w1_kernel(const float* __restrict__ W1, float* __restrict__ Wp) {
    int idx = blockIdx.x * blockDim.x + threadIdx.x;
    if (idx >= KG * 64) return;
    const int g = idx >> 6, r = idx & 63;
    const int l = r >> 1,   j = r & 1;
    const int half = l >> 4, n = l & 15;
    const int k = 4 * g + 2 * half + j;
    Wp[idx] = (k < F_IN) ? W1[k * F_HID + n] : 0.0f;
}

// ---------------- GEMM1: H[N,16] = X[N,1433] @ W1[1433,16] (fp32 WMMA) ----------------
// One wave owns TWO 16-row M-tiles (32 rows): each B fragment feeds 2 WMMAs.
__global__ __launch_bounds__(256) void gemm1_wmma(const float* __restrict__ X,
                                                  const float* __restrict__ Wp,
                                                  float* __restrict__ H, int nNodes) {
    const int lane = threadIdx.x & 31;
    const int wave = blockIdx.x * 8 + (threadIdx.x >> 5);
    const int base = wave * 32;                              // 32 rows per wave
    if (base >= nNodes) return;                              // wave-uniform exit
    const int half = lane >> 4;                              // 0: K0,K1   1: K2,K3
    const int n    = lane & 15;
    const float* __restrict__ xr0 = X + ((long)base + n)      * (long)F_IN;
    const float* __restrict__ xr1 = X + ((long)base + 16 + n) * (long)F_IN;
    const v2f*   __restrict__ wp  = (const v2f*)Wp;

    v8f acc0 = {}, acc1 = {};
    int g = 0;
#pragma unroll 4
    for (; g < KG - 1; ++g) {
        const int ka = 4 * g + 2 * half;
        v2f a0, a1;
        a0.x = __builtin_nontemporal_load(xr0 + ka);         // X streamed once: TH=NT,
        a0.y = __builtin_nontemporal_load(xr0 + ka + 1);     // keep L2 for h1 / edges
        a1.x = __builtin_nontemporal_load(xr1 + ka);
        a1.y = __builtin_nontemporal_load(xr1 + ka + 1);
        const v2f b = wp[g * 32 + lane];                     // one coalesced b64, L2-resident
        acc0 = __builtin_amdgcn_wmma_f32_16x16x4_f32(false, a0, false, b,
                                                     (short)0, acc0, false, false);
        acc1 = __builtin_amdgcn_wmma_f32_16x16x4_f32(false, a1, false, b,
                                                     (short)0, acc1, false, false);
    }
    {   // K tail group: only ka==1432 (half 0, first element) is a valid X column
        const int ka = 4 * g + 2 * half;
        v2f a0 = {0.0f, 0.0f}, a1 = {0.0f, 0.0f};
        if (ka < F_IN) { a0.x = xr0[ka]; a1.x = xr1[ka]; }
        const v2f b = wp[g * 32 + lane];                     // zero-padded: no B predicate
        acc0 = __builtin_amdgcn_wmma_f32_16x16x4_f32(false, a0, false, b,
                                                     (short)0, acc0, false, false);
        acc1 = __builtin_amdgcn_wmma_f32_16x16x4_f32(false, a1, false, b,
                                                     (short)0, acc1, false, false);
    }
    float* __restrict__ hr0 = H + (long)base * F_HID;
    float* __restrict__ hr1 = H + ((long)base + 16) * F_HID;
#pragma unroll
    for (int v = 0; v < 8; ++v) {
        hr0[(v + 8 * half) * F_HID + n] = acc0[v];           // D: M=v+8*half, N=n
        hr1[(v + 8 * half) * F_HID + n] = acc1[v];
    }
}

// ---------------- layer-1 aggregation ----------------
__global__ void selfloop1_kernel(const float* __restrict__ h, const float* __restrict__ dis,
                                 float* __restrict__ a1, int n16) {
    int i = blockIdx.x * blockDim.x + threadIdx.x;
    if (i < n16) {
        float ds = dis[i >> 4];
        a1[i] = h[i] * ds * ds;                              // self-loop message
    }
}

__global__ void edge1_kernel(const int* __restrict__ src, const int* __restrict__ dst,
                             const float* __restrict__ dis, const float* __restrict__ h,
                             float* __restrict__ a1, int e) {
    int i = blockIdx.x * blockDim.x + threadIdx.x;
    if (i >= e) return;
    const int s = src[i], d = dst[i];
    const float w = dis[s] * dis[d];
    const float4* __restrict__ hs = (const float4*)(h + (long)s * F_HID);
    float* __restrict__ ad = a1 + (long)d * F_HID;
#pragma unroll
    for (int j = 0; j < 4; ++j) {
        float4 v = hs[j];
        atomAddF(&ad[4 * j + 0], v.x * w);
        atomAddF(&ad[4 * j + 1], v.y * w);
        atomAddF(&ad[4 * j + 2], v.z * w);
        atomAddF(&ad[4 * j + 3], v.w * w);
    }
}

__global__ void bias_relu_kernel(float* __restrict__ a1, const float* __restrict__ b1, int n16) {
    int i = blockIdx.x * blockDim.x + threadIdx.x;
    if (i < n16) a1[i] = fmaxf(a1[i] + b1[i & 15], 0.0f);
}

// ---------------- GEMM2: H2[N,8pad] = A1[N,16] @ W2[16,7] (fp32 WMMA, N padded to 16) ----
__global__ __launch_bounds__(256) void gemm2_wmma(const float* __restrict__ A,
                                                  const float* __restrict__ W2,
                                                  float* __restrict__ H2, int nNodes) {
    const int lane = threadIdx.x & 31;
    const int tile = blockIdx.x * 8 + (threadIdx.x >> 5);
    if (tile * 16 >= nNodes) return;                        // wave-uniform exit
    const int half = lane >> 4;
    const int n    = lane & 15;
    const long m   = (long)tile * 16 + n;
    const float* __restrict__ ar = A + m * F_HID;
    const int  nc    = (n < F_OUT) ? n : (F_OUT - 1);       // clamped column (select, no EXEC div)
    const bool valid = (n < F_OUT);

    v8f acc = {};
#pragma unroll
    for (int k = 0; k < F_HID; k += 4) {
        const int ka = k + half * 2;
        v2f a; a.x = ar[ka]; a.y = ar[ka + 1];
        float bx = W2[ka * F_OUT + nc];
        float by = W2[(ka + 1) * F_OUT + nc];
        v2f b; b.x = valid ? bx : 0.0f; b.y = valid ? by : 0.0f;
        acc = __builtin_amdgcn_wmma_f32_16x16x4_f32(false, a, false, b,
                                                    (short)0, acc, false, false);
    }
    if (valid) {
        float* __restrict__ hr = H2 + (long)tile * 16 * 8;  // padded stride 8
#pragma unroll
        for (int v = 0; v < 8; ++v)
            hr[(v + 8 * half) * 8 + n] = acc[v];
    }
}

// ---------------- layer-2 aggregation + log_softmax ----------------
__global__ void selfloop2_kernel(const float* __restrict__ h2, const float* __restrict__ dis,
                                 float* __restrict__ out, int n7) {
    int i = blockIdx.x * blockDim.x + threadIdx.x;
    if (i < n7) {
        int node = i / F_OUT, c = i - node * F_OUT;
        float ds = dis[node];
        out[i] = h2[node * 8 + c] * ds * ds;
    }
}

__global__ void edge2_kernel(const int* __restrict__ src, const int* __restrict__ dst,
                             const float* __restrict__ dis, const float* __restrict__ h2,
                             float* __restrict__ out, int e) {
    int i = blockIdx.x * blockDim.x + threadIdx.x;
    if (i >= e) return;
    const int s = src[i], d = dst[i];
    const float w = dis[s] * dis[d];
    const float* __restrict__ hs = h2 + (long)s * 8;
    float* __restrict__ od = out + (long)d * F_OUT;
#pragma unroll
    for (int c = 0; c < F_OUT; ++c) atomAddF(&od[c], hs[c] * w);
}

__global__ void logsoftmax_kernel(float* __restrict__ out, const float* __restrict__ b2, int n) {
    int i = blockIdx.x * blockDim.x + threadIdx.x;
    if (i >= n) return;
    float z[F_OUT];
    float mx = -INFINITY;
#pragma unroll
    for (int c = 0; c < F_OUT; ++c) { z[c] = out[i * F_OUT + c] + b2[c]; mx = fmaxf(mx, z[c]); }
    float s = 0.0f;
#pragma unroll
    for (int c = 0; c < F_OUT; ++c) s += expf(z[c] - mx);
    const float lse = mx + logf(s);
#pragma unroll
    for (int c = 0; c < F_OUT; ++c) out[i * F_OUT + c] = z[c] - lse;
}

// ---------------- launcher ----------------
extern "C" void kernel_launch(void* const* d_in, const int* in_sizes, int n_in,
                              void* d_out, int out_size, void* d_ws, size_t ws_size,
                              hipStream_t stream) {
    const float* x    = (const float*)d_in[0];
    const int*   ei   = (const int*)d_in[1];
    const float* W1   = (const float*)d_in[2];
    const float* b1   = (const float*)d_in[3];
    const float* W2   = (const float*)d_in[4];
    const float* b2   = (const float*)d_in[5];
    float*       out  = (float*)d_out;

    const int N = in_sizes[0] / F_IN;        // 100000
    const int E = in_sizes[1] / 2;           // 3200000
    const int* src = ei;
    const int* dst = ei + E;

    float* ws  = (float*)d_ws;
    float* dis = ws;                         // N
    float* h1  = dis + N;                    // N*16
    float* a1  = h1 + (long)N * F_HID;       // N*16
    float* h2  = a1 + (long)N * F_HID;       // N*8 (padded)
    float* wp  = h2 + (long)N * 8;           // KG*64 packed W1

    const int T = 256;
    const int gN    = (N + T - 1) / T;
    const int gE    = (E + T - 1) / T;
    const int gN16  = (N * F_HID + T - 1) / T;
    const int gN7   = (N * F_OUT + T - 1) / T;
    const int gWp   = (KG * 64 + T - 1) / T;
    const int gG1   = ((N + 31) / 32 + 7) / 8;   // 8 waves x 32 rows per 256-thread block
    const int gG2   = ((N + 15) / 16 + 7) / 8;   // 8 waves x 16 rows per block

    // degrees -> dis ; pack W1 (independent, same stream)
    deg_init_kernel<<<gN, T, 0, stream>>>(dis, N);
    deg_edge_kernel<<<gE, T, 0, stream>>>(dst, dis, E);
    deg_rsqrt_kernel<<<gN, T, 0, stream>>>(dis, N);
    pack_w1_kernel<<<gWp, T, 0, stream>>>(W1, wp);

    // layer 1
    gemm1_wmma<<<gG1, T, 0, stream>>>(x, wp, h1, N);
    selfloop1_kernel<<<gN16, T, 0, stream>>>(h1, dis, a1, N * F_HID);
    edge1_kernel<<<gE, T, 0, stream>>>(src, dst, dis, h1, a1, E);
    bias_relu_kernel<<<gN16, T, 0, stream>>>(a1, b1, N * F_HID);

    // layer 2
    gemm2_wmma<<<gG2, T, 0, stream>>>(a1, W2, h2, N);
    selfloop2_kernel<<<gN7, T, 0, stream>>>(h2, dis, out, N * F_OUT);
    edge2_kernel<<<gE, T, 0, stream>>>(src, dst, dis, h2, out, E);
    logsoftmax_kernel<<<gN, T, 0, stream>>>(out, b2, N);
}